// FiniteScalarCodebook_62440234549487
// MI455X (gfx1250) — compile-verified
//
#include <hip/hip_runtime.h>
#include <hip/hip_bf16.h>
#include <math.h>

typedef __attribute__((ext_vector_type(2))) float v2f;
typedef __attribute__((ext_vector_type(8))) float v8f;

#define NUM_TOKENS 8192
#define NUM_CODES  4096

// Flat float offsets of the tuple outputs in d_out (return order):
// quantize, ste, perplexity, soft_proba, masked_distances, bounded, tokens
#define Q_OFF   0
#define STE_OFF 32768
#define PPL_OFF 65536
#define SP_OFF  65537
#define MD_OFF  (65537 + 33554432)           // 33619969
#define B_OFF   (MD_OFF + 33554432)          // 67174401
#define TOK_OFF (B_OFF + 32768)              // 67207169

// ---------------- workspace zeroing (ws is poisoned 0xAA) ----------------
__global__ void k_zero_ws(unsigned int* __restrict__ ws, int n) {
    int i = blockIdx.x * blockDim.x + threadIdx.x;
    if (i < n) ws[i] = 0u;
}

// ---------------- per-token: bound / quantize / ste / tokens / histogram ----------------
__global__ void k_quantize(const float* __restrict__ in, const float* __restrict__ masks,
                           float* __restrict__ out, unsigned int* __restrict__ hist) {
    int i = blockIdx.x * blockDim.x + threadIdx.x;   // token index
    if (i >= NUM_TOKENS) return;
    const float half_l = 3.4965f;                    // (8-1)*(1-0.001)/2
    const float offset = 0.5f;                       // even levels
    const float shift  = tanf(offset / half_l);
    float tok = 0.0f, basis = 1.0f;
#pragma unroll
    for (int j = 0; j < 4; ++j) {
        float z  = in[i * 4 + j];
        float m  = masks[i * 4 + j];
        float bd = m * (tanhf(z + shift) * half_l - offset);
        float qv = rintf(bd);                        // jnp.round = RNE
        out[Q_OFF   + i * 4 + j] = qv;
        out[STE_OFF + i * 4 + j] = qv;               // ste == quantize numerically
        out[B_OFF   + i * 4 + j] = bd;
        tok  += (qv + 4.0f) * basis;                 // basis = 1,8,64,512
        basis *= 8.0f;
    }
    unsigned int t = (unsigned int)tok;
    out[TOK_OFF + i] = (float)t;
    if (masks[i * 4] > 0.0f) {                       // m = masks[...,0:1]
        atomicAdd(&hist[t], 1u);
        atomicAdd(&hist[NUM_CODES], 1u);             // total
    }
}

// ---------------- perplexity from histogram ----------------
__global__ void k_perplexity(const unsigned int* __restrict__ hist, float* __restrict__ out) {
    __shared__ float red[256];
    int tid = threadIdx.x;
    float total = (float)hist[NUM_CODES];
    float inv   = total > 0.0f ? 1.0f / total : 0.0f;
    float acc = 0.0f;
    for (int c = tid; c < NUM_CODES; c += 256) {
        float p = (float)hist[c] * inv;
        acc += p * logf(p + 1e-10f);
    }
    red[tid] = acc;
    __syncthreads();
    for (int s = 128; s > 0; s >>= 1) {
        if (tid < s) red[tid] += red[tid + s];
        __syncthreads();
    }
    if (tid == 0) out[PPL_OFF] = expf(-red[0]);
}

// ---------------- one 16x16 distance tile via V_WMMA_F32_16X16X4_F32 ----------------
// A (16x4) = bounded rows, B (4x16) = codebook^T synthesized from code bits.
__device__ __forceinline__ void tile_dist(v2f a, int c0, int lane, const float* bn, float* dist) {
    int n  = lane & 15;
    int c  = c0 + n;
    int k0 = (lane < 16) ? 0 : 2;                    // ISA 16x4 A / 4x16 B lane layout
    v2f b;
    b.x = (float)(((c >> (3 * k0))     & 7) - 4);
    b.y = (float)(((c >> (3 * k0 + 3)) & 7) - 4);
    float e0 = (float)(( c        & 7) - 4);
    float e1 = (float)(((c >> 3)  & 7) - 4);
    float e2 = (float)(((c >> 6)  & 7) - 4);
    float e3 = (float)(((c >> 9)  & 7) - 4);
    float cn = e0 * e0 + e1 * e1 + e2 * e2 + e3 * e3;
    v8f acc = {0.f, 0.f, 0.f, 0.f, 0.f, 0.f, 0.f, 0.f};
    // (neg_a, A, neg_b, B, c_mod, C, reuse_a, reuse_b)
    acc = __builtin_amdgcn_wmma_f32_16x16x4_f32(false, a, false, b, (short)0, acc, false, false);
#pragma unroll
    for (int r = 0; r < 8; ++r) dist[r] = bn[r] + cn - 2.0f * acc[r];
}

// ---------------- distances + softmax: 16 tokens/block, 8 waves x 32 tiles ----------------
__global__ void __launch_bounds__(256) k_dist_softmax(const float* __restrict__ bounded_g,
                                                      const float* __restrict__ masks,
                                                      float* __restrict__ out) {
    __shared__ float s_b[16][4];
    __shared__ float s_bn[16];
    __shared__ float s_m[16];
    __shared__ float s_wstat[8][16];
    __shared__ float s_rstat[16];

    int tid  = threadIdx.x;
    int lane = tid & 31;
    int wave = tid >> 5;
    int row0 = blockIdx.x * 16;

    if (tid < 16) {
        float b0 = bounded_g[(row0 + tid) * 4 + 0];
        float b1 = bounded_g[(row0 + tid) * 4 + 1];
        float b2 = bounded_g[(row0 + tid) * 4 + 2];
        float b3 = bounded_g[(row0 + tid) * 4 + 3];
        s_b[tid][0] = b0; s_b[tid][1] = b1; s_b[tid][2] = b2; s_b[tid][3] = b3;
        s_bn[tid] = b0 * b0 + b1 * b1 + b2 * b2 + b3 * b3;
        s_m[tid]  = masks[(row0 + tid) * 4];
    }
    __syncthreads();

    int tok = lane & 15;
    int k0  = (lane < 16) ? 0 : 2;
    v2f a;
    a.x = s_b[tok][k0];
    a.y = s_b[tok][k0 + 1];

    int half8 = (lane < 16) ? 0 : 8;                 // D-layout: VGPR r -> row r (+8 for hi half)
    float bn[8], mr[8];
#pragma unroll
    for (int r = 0; r < 8; ++r) { bn[r] = s_bn[r + half8]; mr[r] = s_m[r + half8]; }

    float* out_md = out + MD_OFF;
    float* out_sp = out + SP_OFF;

    // ---- Pass 1: masked distances (NT store) + running row max ----
    float vmax[8];
#pragma unroll
    for (int r = 0; r < 8; ++r) vmax[r] = -3.0e38f;
    for (int t = 0; t < 32; ++t) {
        int c0 = wave * 512 + t * 16;
        float dist[8];
        tile_dist(a, c0, lane, bn, dist);
        int c = c0 + (lane & 15);
#pragma unroll
        for (int r = 0; r < 8; ++r) {
            vmax[r] = fmaxf(vmax[r], dist[r]);
            size_t idx = (size_t)(row0 + r + half8) * NUM_CODES + c;
            __builtin_nontemporal_store(mr[r] * dist[r], &out_md[idx]);
        }
    }
#pragma unroll
    for (int r = 0; r < 8; ++r)
#pragma unroll
        for (int s = 1; s < 16; s <<= 1)
            vmax[r] = fmaxf(vmax[r], __shfl_xor(vmax[r], s, 32));
    if ((lane & 15) == 0)
#pragma unroll
        for (int r = 0; r < 8; ++r) s_wstat[wave][r + half8] = vmax[r];
    __syncthreads();
    if (tid < 16) {
        float mx = s_wstat[0][tid];
#pragma unroll
        for (int w = 1; w < 8; ++w) mx = fmaxf(mx, s_wstat[w][tid]);
        s_rstat[tid] = mx;
    }
    __syncthreads();
    float rmax[8];
#pragma unroll
    for (int r = 0; r < 8; ++r) rmax[r] = s_rstat[r + half8];
    __syncthreads();

    // ---- Pass 2: recompute tiles, accumulate sum of exp ----
    float vsum[8];
#pragma unroll
    for (int r = 0; r < 8; ++r) vsum[r] = 0.0f;
    for (int t = 0; t < 32; ++t) {
        int c0 = wave * 512 + t * 16;
        float dist[8];
        tile_dist(a, c0, lane, bn, dist);
#pragma unroll
        for (int r = 0; r < 8; ++r) vsum[r] += __expf(dist[r] - rmax[r]);
    }
#pragma unroll
    for (int r = 0; r < 8; ++r)
#pragma unroll
        for (int s = 1; s < 16; s <<= 1)
            vsum[r] += __shfl_xor(vsum[r], s, 32);
    if ((lane & 15) == 0)
#pragma unroll
        for (int r = 0; r < 8; ++r) s_wstat[wave][r + half8] = vsum[r];
    __syncthreads();
    if (tid < 16) {
        float sm = 0.0f;
#pragma unroll
        for (int w = 0; w < 8; ++w) sm += s_wstat[w][tid];
        s_rstat[tid] = 1.0f / sm;
    }
    __syncthreads();
    float rinv[8];
#pragma unroll
    for (int r = 0; r < 8; ++r) rinv[r] = s_rstat[r + half8];

    // ---- Pass 3: recompute tiles, write normalized softmax (NT store) ----
    for (int t = 0; t < 32; ++t) {
        int c0 = wave * 512 + t * 16;
        float dist[8];
        tile_dist(a, c0, lane, bn, dist);
        int c = c0 + (lane & 15);
#pragma unroll
        for (int r = 0; r < 8; ++r) {
            size_t idx = (size_t)(row0 + r + half8) * NUM_CODES + c;
            __builtin_nontemporal_store(__expf(dist[r] - rmax[r]) * rinv[r], &out_sp[idx]);
        }
    }
}

extern "C" void kernel_launch(void* const* d_in, const int* in_sizes, int n_in,
                              void* d_out, int out_size, void* d_ws, size_t ws_size,
                              hipStream_t stream) {
    const float* in  = (const float*)d_in[0];   // inputs (8,1024,4)
    const float* mk  = (const float*)d_in[1];   // masks  (8,1024,4)
    float* out       = (float*)d_out;
    unsigned int* hist = (unsigned int*)d_ws;   // 4096 bins + total

    k_zero_ws<<<(NUM_CODES + 1 + 255) / 256, 256, 0, stream>>>(hist, NUM_CODES + 1);
    k_quantize<<<NUM_TOKENS / 256, 256, 0, stream>>>(in, mk, out, hist);
    k_perplexity<<<1, 256, 0, stream>>>(hist, out);
    k_dist_softmax<<<NUM_TOKENS / 16, 256, 0, stream>>>(out + B_OFF, mk, out);
}